// Co_Location_Loss_59785944761050
// MI455X (gfx1250) — compile-verified
//
#include <hip/hip_runtime.h>
#include <hip/hip_bf16.h>

// ---------------------------------------------------------------------------
// Co-location loss on MI455X (gfx1250), f32 WMMA 16x16x4 path.
//
// loss = sum_{b,k<valid[b],h,w} | sum_c Pdiff[b,k,c] * V[b,c,h,w] | / C
//        / sum_b valid[b]
// with Pdiff[b,k,c] = obj[b,k,:]·W1[c,:] - attr[b,k,:]·W2[c,:] + b1[c]-b2[c]
// (legal because |.| is applied after the channel contraction).
//
// Roofline: V = 151 MB dominates (>= ~7 us at 23.3 TB/s); everything else is
// L2-resident (192 MB). Fusing obj/attr into one Pdiff halves the V-side
// FLOPs; each 16-wide HW tile owns all 2048 channels so V is read once.
// ---------------------------------------------------------------------------

typedef __attribute__((ext_vector_type(2))) float v2f;
typedef __attribute__((ext_vector_type(8))) float v8f;

#define IMG_C 2048
#define EMB_D 1024
#define NSLOT 20
#define MPAD  32      // slot rows padded to 2 WMMA M-tiles
#define BATCH 32
#define SEQL  100
#define HW    576     // 24*24

#define ACT_WAVES 4               // K-split waves per act block
#define CCHUNK   (IMG_C / ACT_WAVES)
#define LDS_STRIDE 17             // 16 values + 1 pad -> conflict-free banks

static __device__ __forceinline__ v8f wmma_f32_k4(v2f a, v2f b, v8f c) {
  // D = A(16x4) * B(4x16) + C(16x16), all f32.
  return __builtin_amdgcn_wmma_f32_16x16x4_f32(
      /*neg_a=*/false, a, /*neg_b=*/false, b,
      /*c_mod=*/(short)0, c, /*reuse_a=*/false, /*reuse_b=*/false);
}

// ---------------------------------------------------------------------------
// Kernel 0: zero the global accumulator in workspace.
// ---------------------------------------------------------------------------
__global__ void __launch_bounds__(32)
init_kernel(float* __restrict__ sum) {
  if (threadIdx.x == 0) sum[0] = 0.0f;
}

// ---------------------------------------------------------------------------
// Kernel 1: Pdiff[b, m(32), c] via doubled-K GEMM.
//   A' (32 x 2048) = [ obj_emb | -attr_emb ]    (rows >= 20 are don't-care)
//   B' (2048 x 16) = [ W1ᵀ ; W2ᵀ ] tile of 16 channels
// One wave per (batch, channel-tile); two M-tile accumulators share B frags.
// ---------------------------------------------------------------------------
__global__ void __launch_bounds__(32)
proj_kernel(const float* __restrict__ S,   // [B, SEQL, EMB_D]
            const float* __restrict__ W1,  // [IMG_C, EMB_D]
            const float* __restrict__ b1,  // [IMG_C]
            const float* __restrict__ W2,  // [IMG_C, EMB_D]
            const float* __restrict__ b2,  // [IMG_C]
            const int*   __restrict__ oidx,// [B, NSLOT]
            const int*   __restrict__ aidx,// [B, NSLOT]
            float*       __restrict__ Pd)  // [B, MPAD, IMG_C]
{
  const int lane  = (int)threadIdx.x;
  const int b     = (int)blockIdx.y;
  const int ctile = (int)blockIdx.x;
  const int nlo   = lane & 15;
  const int khalf = (lane >> 4) << 1;          // 0 or 2
  const int col   = ctile * 16 + nlo;          // output channel

  // A-matrix row pointers (per ISA A-layout: row = lane&15 [+16]).
  const int r0 = nlo;
  const int r1 = (nlo + 16 < NSLOT) ? (nlo + 16) : 0;  // clamp padded rows (finite junk, masked later)
  const float* po0 = S + (size_t)(b * SEQL + oidx[b * NSLOT + r0]) * EMB_D;
  const float* pa0 = S + (size_t)(b * SEQL + aidx[b * NSLOT + r0]) * EMB_D;
  const float* po1 = S + (size_t)(b * SEQL + oidx[b * NSLOT + r1]) * EMB_D;
  const float* pa1 = S + (size_t)(b * SEQL + aidx[b * NSLOT + r1]) * EMB_D;
  const float* pw1 = W1 + (size_t)col * EMB_D; // B' frag: W[col][d], d contiguous
  const float* pw2 = W2 + (size_t)col * EMB_D;

  v8f acc0 = {0.f,0.f,0.f,0.f,0.f,0.f,0.f,0.f};
  v8f acc1 = {0.f,0.f,0.f,0.f,0.f,0.f,0.f,0.f};

  // First K-half: +obj x W1
  for (int k = 0; k < EMB_D; k += 4) {
    const int d = k + khalf;
    v2f a0 = *(const v2f*)(po0 + d);
    v2f a1 = *(const v2f*)(po1 + d);
    v2f bb = *(const v2f*)(pw1 + d);
    acc0 = wmma_f32_k4(a0, bb, acc0);
    acc1 = wmma_f32_k4(a1, bb, acc1);
  }
  // Second K-half: -attr x W2 (f32 WMMA has no A-negate modifier; negate in VALU)
  for (int k = 0; k < EMB_D; k += 4) {
    const int d = k + khalf;
    v2f l0 = *(const v2f*)(pa0 + d);
    v2f l1 = *(const v2f*)(pa1 + d);
    v2f a0 = { -l0.x, -l0.y };
    v2f a1 = { -l1.x, -l1.y };
    v2f bb = *(const v2f*)(pw2 + d);
    acc0 = wmma_f32_k4(a0, bb, acc0);
    acc1 = wmma_f32_k4(a1, bb, acc1);
  }

  // Epilogue: bias diff (same column for all 8 rows of a lane), store Pdiff.
  const float bd    = b1[col] - b2[col];
  const int   mbase = (lane < 16) ? 0 : 8;    // C/D layout: VGPR g -> M = g (+8)
  float* outp = Pd + (size_t)b * MPAD * IMG_C + col;
#pragma unroll
  for (int g = 0; g < 8; ++g) {
    const int m = mbase + g;
    outp[(size_t)m * IMG_C]        = acc0[g] + bd;   // rows 0..15
    outp[(size_t)(m + 16) * IMG_C] = acc1[g] + bd;   // rows 16..31
  }
}

// ---------------------------------------------------------------------------
// Kernel 2: Act[b, m, hw] = sum_c Pdiff[b,m,c] * V[b,c,hw], then masked
// |.|/C reduction. This kernel moves the 151 MB feature map, so occupancy
// matters most: 4 waves per block split the 2048-channel contraction into
// 512-channel chunks (4608 waves in flight vs 1152), partials meet in LDS
// (|.| is only legal after the FULL channel sum), wave 0 finishes.
// ---------------------------------------------------------------------------
__global__ void __launch_bounds__(32 * ACT_WAVES)
act_kernel(const float* __restrict__ V,    // [B, IMG_C, HW]
           const float* __restrict__ Pd,   // [B, MPAD, IMG_C]
           const int*   __restrict__ oidx, // [B, NSLOT]
           const int*   __restrict__ aidx, // [B, NSLOT]
           float*       __restrict__ sum_out)
{
  __shared__ float part[ACT_WAVES * 32 * LDS_STRIDE];

  const int tid   = (int)threadIdx.x;
  const int lane  = tid & 31;
  const int wave  = tid >> 5;
  const int b     = (int)blockIdx.y;
  const int nt    = (int)blockIdx.x;
  const int nlo   = lane & 15;
  const int khalf = (lane >> 4) << 1;          // 0 or 2
  const int col   = nt * 16 + nlo;             // hw index (576 = 36 tiles exactly)

  const float* Vb = V  + (size_t)b * IMG_C * HW;
  const float* P0 = Pd + (size_t)b * MPAD * IMG_C + (size_t)nlo * IMG_C;
  const float* P1 = P0 + (size_t)16 * IMG_C;

  v8f acc0 = {0.f,0.f,0.f,0.f,0.f,0.f,0.f,0.f};
  v8f acc1 = {0.f,0.f,0.f,0.f,0.f,0.f,0.f,0.f};

  const int cbeg = wave * CCHUNK;
  const int cend = cbeg + CCHUNK;
  for (int c = cbeg; c < cend; c += 4) {
    const int c0 = c + khalf;
    v2f a0 = *(const v2f*)(P0 + c0);           // A: Pdiff rows, c contiguous
    v2f a1 = *(const v2f*)(P1 + c0);
    v2f bb;                                    // B: V[c][hw], stride HW between k's
    bb.x = Vb[(size_t)c0 * HW + col];
    bb.y = Vb[(size_t)(c0 + 1) * HW + col];
    acc0 = wmma_f32_k4(a0, bb, acc0);
    acc1 = wmma_f32_k4(a1, bb, acc1);
  }

  // Stage this wave's 16 per-lane partials in LDS (stride-17 pad: bank-clean).
  float* myp = &part[(wave * 32 + lane) * LDS_STRIDE];
#pragma unroll
  for (int g = 0; g < 8; ++g) {
    myp[g]     = acc0[g];
    myp[8 + g] = acc1[g];
  }
  __syncthreads();

  if (wave == 0) {
    // Sum the K-chunks: full contraction now complete per (m, hw) element.
    float v0[8], v1[8];
#pragma unroll
    for (int g = 0; g < 8; ++g) {
      float s0 = 0.f, s1 = 0.f;
#pragma unroll
      for (int w = 0; w < ACT_WAVES; ++w) {
        const float* p = &part[(w * 32 + lane) * LDS_STRIDE];
        s0 += p[g];
        s1 += p[8 + g];
      }
      v0[g] = s0;
      v1[g] = s1;
    }

    // valid[b] = #(obj_idx != attr_idx); slot mask keeps k < valid (<= 20, so
    // the 12 padded rows self-mask).
    int valid = 0;
#pragma unroll
    for (int k = 0; k < NSLOT; ++k)
      valid += (oidx[b * NSLOT + k] != aidx[b * NSLOT + k]) ? 1 : 0;

    const float inv_c = 1.0f / (float)IMG_C;
    const int   mbase = (lane < 16) ? 0 : 8;   // C/D layout: VGPR g -> M = g (+8)
    float s = 0.0f;
#pragma unroll
    for (int g = 0; g < 8; ++g) {
      const int m0 = mbase + g;
      if (m0      < valid) s += __builtin_fabsf(v0[g]);
      if (m0 + 16 < valid) s += __builtin_fabsf(v1[g]);
    }
    s *= inv_c;

    // wave32 butterfly reduction, then one global f32 atomic per block.
#pragma unroll
    for (int off = 16; off > 0; off >>= 1)
      s += __shfl_xor(s, off, 32);
    if (lane == 0) atomicAdd(sum_out, s);
  }
}

// ---------------------------------------------------------------------------
// Kernel 3: divide by total valid count. One wave, one batch per lane.
// ---------------------------------------------------------------------------
__global__ void __launch_bounds__(32)
finalize_kernel(const float* __restrict__ sum_in,
                const int*   __restrict__ oidx,
                const int*   __restrict__ aidx,
                float*       __restrict__ out)
{
  const int lane = (int)threadIdx.x;   // 32 lanes == BATCH
  int v = 0;
#pragma unroll
  for (int k = 0; k < NSLOT; ++k)
    v += (oidx[lane * NSLOT + k] != aidx[lane * NSLOT + k]) ? 1 : 0;
#pragma unroll
  for (int off = 16; off > 0; off >>= 1)
    v += __shfl_xor(v, off, 32);
  if (lane == 0) out[0] = sum_in[0] / (float)v;
}

// ---------------------------------------------------------------------------
// Launch. Inputs (setup_inputs order):
//   0: feature_map_v [32,2048,24,24] f32      4: W2 [2048,1024] f32
//   1: sentence_embedding_s [32,100,1024] f32 5: b2 [2048] f32
//   2: W1 [2048,1024] f32                     6: object_idx [32,20] i32
//   3: b1 [2048] f32                          7: attribute_idx [32,20] i32
// Workspace: [0..3] f32 accumulator, Pdiff at +256 bytes (8 MB).
// ---------------------------------------------------------------------------
extern "C" void kernel_launch(void* const* d_in, const int* in_sizes, int n_in,
                              void* d_out, int out_size, void* d_ws, size_t ws_size,
                              hipStream_t stream) {
  const float* V    = (const float*)d_in[0];
  const float* S    = (const float*)d_in[1];
  const float* W1   = (const float*)d_in[2];
  const float* b1   = (const float*)d_in[3];
  const float* W2   = (const float*)d_in[4];
  const float* b2   = (const float*)d_in[5];
  const int*   oidx = (const int*)d_in[6];
  const int*   aidx = (const int*)d_in[7];
  float* out = (float*)d_out;

  float* sum = (float*)d_ws;
  float* Pd  = (float*)((char*)d_ws + 256);   // [32][32][2048] f32 = 8 MB

  init_kernel<<<dim3(1), dim3(32), 0, stream>>>(sum);
  proj_kernel<<<dim3(IMG_C / 16, BATCH), dim3(32), 0, stream>>>(
      S, W1, b1, W2, b2, oidx, aidx, Pd);
  act_kernel<<<dim3(HW / 16, BATCH), dim3(32 * ACT_WAVES), 0, stream>>>(
      V, Pd, oidx, aidx, sum);
  finalize_kernel<<<dim3(1), dim3(32), 0, stream>>>(sum, oidx, aidx, out);
}